// Memory_79929341378796
// MI455X (gfx1250) — compile-verified
//
#include <hip/hip_runtime.h>
#include <math.h>

typedef float v2f __attribute__((ext_vector_type(2)));
typedef float v8f __attribute__((ext_vector_type(8)));

#ifndef EPSF
#define EPSF 1e-8f
#endif

// Problem constants (from setup_inputs): B=16, N=2048, W=64, R=4
static constexpr int cB = 16;
static constexpr int cN = 2048;
static constexpr int cW = 64;
static constexpr int cR = 4;

// ---------------------------------------------------------------------------
// block-wide reductions (all threads participate; blockDim.x threads)
// ---------------------------------------------------------------------------
__device__ __forceinline__ float blockReduceMax(float v, float* red) {
    const int tid = threadIdx.x;
    red[tid] = v;
    __syncthreads();
    for (int s = blockDim.x >> 1; s > 0; s >>= 1) {
        if (tid < s) red[tid] = fmaxf(red[tid], red[tid + s]);
        __syncthreads();
    }
    float r = red[0];
    __syncthreads();
    return r;
}

__device__ __forceinline__ float blockReduceSum(float v, float* red) {
    const int tid = threadIdx.x;
    red[tid] = v;
    __syncthreads();
    for (int s = blockDim.x >> 1; s > 0; s >>= 1) {
        if (tid < s) red[tid] += red[tid + s];
        __syncthreads();
    }
    float r = red[0];
    __syncthreads();
    return r;
}

// ---------------------------------------------------------------------------
// K1: allocation weight (sort-free, equivalent to stable argsort + cumprod)
// alloc[i] = (1-u[i]) * prod_{j : u[j]<u[i] || (u[j]==u[i] && j<i)} u[j]
// grid = B, block = 1024
// ---------------------------------------------------------------------------
__global__ __launch_bounds__(1024) void k_alloc(const float* __restrict__ usage,
                                                float* __restrict__ ws_alloc) {
    __shared__ float su[cN];
    const int b = blockIdx.x, tid = threadIdx.x;
    su[tid]        = usage[b * cN + tid];
    su[tid + 1024] = usage[b * cN + tid + 1024];
    __syncthreads();
    const int i0 = tid, i1 = tid + 1024;
    const float u0 = su[i0], u1 = su[i1];
    float p0 = 1.f, p1 = 1.f;
    for (int j = 0; j < cN; ++j) {
        const float uj = su[j];
        p0 *= ((uj < u0) || ((uj == u0) && (j < i0))) ? uj : 1.f;
        p1 *= ((uj < u1) || ((uj == u1) && (j < i1))) ? uj : 1.f;
    }
    ws_alloc[b * cN + i0] = (1.f - u0) * p0;
    ws_alloc[b * cN + i1] = (1.f - u1) * p1;
}

// ---------------------------------------------------------------------------
// K2: write content addressing (old memory) + ww + per-batch sum(ww)
// grid = B, block = 1024
// ---------------------------------------------------------------------------
__global__ __launch_bounds__(1024) void k_wcontent(
    const float* __restrict__ wkey, const float* __restrict__ wstr,
    const float* __restrict__ agate, const float* __restrict__ wgate,
    const float* __restrict__ memory, const float* __restrict__ ws_alloc,
    float* __restrict__ o_ww, float* __restrict__ ws_wwsum) {
    __shared__ float red[1024];
    const int b = blockIdx.x, tid = threadIdx.x;

    // write-key norm (redundant per thread; 64 elems)
    const float4* kp = (const float4*)(wkey + (size_t)b * cW);
    float kk = 0.f;
#pragma unroll
    for (int q = 0; q < 16; ++q) {
        float4 k4 = kp[q];
        kk += k4.x * k4.x + k4.y * k4.y + k4.z * k4.z + k4.w * k4.w;
    }
    const float kn  = fmaxf(sqrtf(kk), EPSF);
    const float str = wstr[b];
    const float ag  = agate[b];
    const float wg  = wgate[b];

    const int n0 = tid, n1 = tid + 1024;
    float sim[2];
#pragma unroll
    for (int k = 0; k < 2; ++k) {
        const int nn = (k == 0) ? n0 : n1;
        const float4* mp = (const float4*)(memory + ((size_t)b * cN + nn) * cW);
        float d = 0.f, mm = 0.f;
#pragma unroll
        for (int q = 0; q < 16; ++q) {
            float4 m4 = mp[q];
            float4 k4 = kp[q];
            d  += k4.x * m4.x + k4.y * m4.y + k4.z * m4.z + k4.w * m4.w;
            mm += m4.x * m4.x + m4.y * m4.y + m4.z * m4.z + m4.w * m4.w;
        }
        const float mn = fmaxf(sqrtf(mm), EPSF);
        sim[k] = (d / (kn * mn)) * str;
    }
    const float mx = blockReduceMax(fmaxf(sim[0], sim[1]), red);
    const float e0 = __expf(sim[0] - mx);
    const float e1 = __expf(sim[1] - mx);
    const float tot = blockReduceSum(e0 + e1, red);

    float wwloc = 0.f;
#pragma unroll
    for (int k = 0; k < 2; ++k) {
        const int nn = (k == 0) ? n0 : n1;
        const float wc = ((k == 0) ? e0 : e1) / tot;
        const float al = ws_alloc[b * cN + nn];
        const float w  = wg * (ag * (al - wc) + wc);
        o_ww[b * cN + nn] = w;
        wwloc += w;
    }
    const float wsum = blockReduceSum(wwloc, red);
    if (tid == 0) ws_wwsum[b] = wsum;
}

// ---------------------------------------------------------------------------
// K3: memory erase/write update.  grid*block covers B*N*16 float4s.
// ---------------------------------------------------------------------------
__global__ __launch_bounds__(256) void k_memupd(
    const float* __restrict__ memory, const float* __restrict__ erase,
    const float* __restrict__ wvec, const float* __restrict__ wwv,
    float* __restrict__ o_mem) {
    const int idx = blockIdx.x * 256 + threadIdx.x;   // B*N*16 = 524288
    const int b   = idx >> 15;
    const int rem = idx & 32767;
    const int nn  = rem >> 4;
    const int wq  = rem & 15;
    const float w = wwv[b * cN + nn];
    const float4 m4 = *(const float4*)(memory + ((size_t)b * cN + nn) * cW + wq * 4);
    const float4 e4 = *(const float4*)(erase + (size_t)b * cW + wq * 4);
    const float4 v4 = *(const float4*)(wvec + (size_t)b * cW + wq * 4);
    float4 o;
    o.x = m4.x * (1.f - w * e4.x) + w * v4.x;
    o.y = m4.y * (1.f - w * e4.y) + w * v4.y;
    o.z = m4.z * (1.f - w * e4.z) + w * v4.z;
    o.w = m4.w * (1.f - w * e4.w) + w * v4.w;
    *(float4*)(o_mem + ((size_t)b * cN + nn) * cW + wq * 4) = o;
}

// ---------------------------------------------------------------------------
// K4: fused link update + fwd/bwd matmul partials via V_WMMA_F32_16X16X4_F32.
// grid = B*16*16 (128x128 tiles), block = 256 (8 waves; wave w owns 16 i-rows)
//   link[i,j] = (1 - ww[i] - ww[j]) * L[i,j] + ww[i]*prec[j], diag zeroed
//   fwd_part[b,jb,r,i] = sum_{j in jb} rw[r,j] * link[i,j]
//   bwd_part[b,ib,r,j] = sum_{i in ib} rw[r,i] * link[i,j]
// ---------------------------------------------------------------------------
__global__ __launch_bounds__(256) void k_link(
    const float* __restrict__ link_in, const float* __restrict__ prec_in,
    const float* __restrict__ rw_in, const float* __restrict__ ww,
    float* __restrict__ link_out, float* __restrict__ fwd_part,
    float* __restrict__ bwd_part) {
    // per-wave 16x64 link strip, padded stride 68 (bank-spread, 16B aligned)
    __shared__ float lds_link[8 * 16 * 68];   // 34.8 KB
    __shared__ float lds_rwj[16 * 130];       // rw over j-block, rows 4..15 = 0
    __shared__ float lds_rwi[16 * 130];       // rw over i-block, rows 4..15 = 0
    __shared__ float lds_bwd[cR * 128];       // bwd accumulation for this tile

    const int tid  = threadIdx.x;
    const int wv   = tid >> 5;
    const int lane = tid & 31;
    const int bx   = blockIdx.x;
    const int b    = bx >> 8;
    const int rem  = bx & 255;
    const int ib   = rem >> 4;
    const int jb   = rem & 15;
    const int i0   = ib * 128 + wv * 16;
    const int j0   = jb * 128;

    for (int t = tid; t < 16 * 128; t += 256) {
        const int r = t >> 7, c = t & 127;
        const float vj = (r < cR) ? rw_in[(size_t)(b * cR + r) * cN + j0 + c] : 0.f;
        const float vi = (r < cR) ? rw_in[(size_t)(b * cR + r) * cN + ib * 128 + c] : 0.f;
        lds_rwj[r * 130 + c] = vj;
        lds_rwi[r * 130 + c] = vi;
    }
    for (int t = tid; t < cR * 128; t += 256) lds_bwd[t] = 0.f;
    __syncthreads();

    float* myl = &lds_link[wv * 16 * 68];
    const int lrow = lane >> 4;          // 0/1: K-pair half
    const int lidx = lane & 15;          // M or N index
    const int lcol = lidx * 4;

    v8f dfwd = {0.f, 0.f, 0.f, 0.f, 0.f, 0.f, 0.f, 0.f};
    const size_t lbase = (size_t)b * cN * cN;

    for (int half = 0; half < 2; ++half) {
        const int jh = j0 + half * 64;
        // -------- load L, compute link, store to global + LDS --------
#pragma unroll
        for (int m = 0; m < 8; ++m) {
            const int row = 2 * m + lrow;
            const int gi  = i0 + row;
            const float4 old4 = *(const float4*)(link_in + lbase + (size_t)gi * cN + jh + lcol);
            const float  wwi  = ww[b * cN + gi];
            const float4 wwj4 = *(const float4*)(ww + b * cN + jh + lcol);
            const float4 pj4  = *(const float4*)(prec_in + b * cN + jh + lcol);
            float4 lv;
            lv.x = (gi == jh + lcol + 0) ? 0.f : (1.f - wwi - wwj4.x) * old4.x + wwi * pj4.x;
            lv.y = (gi == jh + lcol + 1) ? 0.f : (1.f - wwi - wwj4.y) * old4.y + wwi * pj4.y;
            lv.z = (gi == jh + lcol + 2) ? 0.f : (1.f - wwi - wwj4.z) * old4.z + wwi * pj4.z;
            lv.w = (gi == jh + lcol + 3) ? 0.f : (1.f - wwi - wwj4.w) * old4.w + wwi * pj4.w;
            *(float4*)(link_out + lbase + (size_t)gi * cN + jh + lcol) = lv;
            *(float4*)(myl + row * 68 + lcol) = lv;
        }
        if (half == 0) {
            __builtin_prefetch(link_in + lbase + (size_t)i0 * cN + j0 + 64, 0, 1);
        }
        __syncthreads();

        // -------- FWD: D(i,r) += A(link 16x4) * B(rw^T 4x16), K-sweep over 64 j
        for (int t = 0; t < 16; ++t) {
            const int kb = t * 4 + 2 * lrow;
            v2f a  = *(const v2f*)(myl + lidx * 68 + kb);
            v2f bb = *(const v2f*)(&lds_rwj[lidx * 130 + half * 64 + kb]);
            dfwd = __builtin_amdgcn_wmma_f32_16x16x4_f32(false, a, false, bb,
                                                         (short)0, dfwd, false, false);
        }
        // -------- BWD: D(j,r) += A(link^T 16x4) * B(rw^T 4x16), K over 16 i
        for (int jt = 0; jt < 4; ++jt) {
            v8f dbwd = {0.f, 0.f, 0.f, 0.f, 0.f, 0.f, 0.f, 0.f};
            const int jl = jt * 16 + lidx;
            for (int t = 0; t < 4; ++t) {
                const int il = t * 4 + 2 * lrow;
                v2f a;
                a.x = myl[il * 68 + jl];
                a.y = myl[(il + 1) * 68 + jl];
                v2f bb = *(const v2f*)(&lds_rwi[lidx * 130 + wv * 16 + il]);
                dbwd = __builtin_amdgcn_wmma_f32_16x16x4_f32(false, a, false, bb,
                                                             (short)0, dbwd, false, false);
            }
            if (lidx < cR) {
#pragma unroll
                for (int v = 0; v < 8; ++v) {
                    const int jloc = half * 64 + jt * 16 + v + 8 * lrow;
                    atomicAdd(&lds_bwd[lidx * 128 + jloc], dbwd[v]);
                }
            }
        }
        __syncthreads();
    }

    // fwd partials: exact (this block owns (i in ib-block) x (jb) fully)
    if (lidx < cR) {
#pragma unroll
        for (int v = 0; v < 8; ++v) {
            const int gi = i0 + v + 8 * lrow;
            fwd_part[((size_t)(b * 16 + jb) * cR + lidx) * cN + gi] = dfwd[v];
        }
    }
    __syncthreads();
    for (int t = tid; t < cR * 128; t += 256) {
        const int r = t >> 7, c = t & 127;
        bwd_part[((size_t)(b * 16 + ib) * cR + r) * cN + j0 + c] = lds_bwd[t];
    }
}

// ---------------------------------------------------------------------------
// K5: read content addressing (new mem) + combine rw + prec + usage_new
// grid = B, block = 1024
// ---------------------------------------------------------------------------
__global__ __launch_bounds__(1024) void k_read(
    const float* __restrict__ rkeys, const float* __restrict__ rstr,
    const float* __restrict__ rmodes, const float* __restrict__ fgates,
    const float* __restrict__ usage_in, const float* __restrict__ prec_in,
    const float* __restrict__ mem, const float* __restrict__ wwv,
    const float* __restrict__ wwsum, const float* __restrict__ fwd_part,
    const float* __restrict__ bwd_part, float* __restrict__ o_rw,
    float* __restrict__ o_prec, float* __restrict__ o_usage) {
    __shared__ float red[1024];
    const int b = blockIdx.x, tid = threadIdx.x;

    float kn[cR], str[cR], m0[cR], m1[cR], m2[cR], fg[cR];
#pragma unroll
    for (int r = 0; r < cR; ++r) {
        const float4* kp = (const float4*)(rkeys + (size_t)(b * cR + r) * cW);
        float s = 0.f;
#pragma unroll
        for (int q = 0; q < 16; ++q) {
            float4 k4 = kp[q];
            s += k4.x * k4.x + k4.y * k4.y + k4.z * k4.z + k4.w * k4.w;
        }
        kn[r]  = fmaxf(sqrtf(s), EPSF);
        str[r] = rstr[b * cR + r];
        m0[r]  = rmodes[(b * cR + r) * 3 + 0];
        m1[r]  = rmodes[(b * cR + r) * 3 + 1];
        m2[r]  = rmodes[(b * cR + r) * 3 + 2];
        fg[r]  = fgates[b * cR + r];
    }
    float S = 0.f;
#pragma unroll
    for (int i = 0; i < cB; ++i) S += wwsum[i];

    const int nidx[2] = {tid, tid + 1024};
    float sim[cR][2];
#pragma unroll
    for (int k = 0; k < 2; ++k) {
        const int nn = nidx[k];
        const float4* mp = (const float4*)(mem + ((size_t)b * cN + nn) * cW);
        float d[cR] = {0.f, 0.f, 0.f, 0.f};
        float mm = 0.f;
#pragma unroll
        for (int q = 0; q < 16; ++q) {
            float4 m4 = mp[q];
            mm += m4.x * m4.x + m4.y * m4.y + m4.z * m4.z + m4.w * m4.w;
#pragma unroll
            for (int r = 0; r < cR; ++r) {
                float4 k4 = ((const float4*)(rkeys + (size_t)(b * cR + r) * cW))[q];
                d[r] += k4.x * m4.x + k4.y * m4.y + k4.z * m4.z + k4.w * m4.w;
            }
        }
        const float mn = fmaxf(sqrtf(mm), EPSF);
#pragma unroll
        for (int r = 0; r < cR; ++r) sim[r][k] = (d[r] / (kn[r] * mn)) * str[r];
        const float wloc = wwv[b * cN + nn];
        o_prec[b * cN + nn] = (1.f - S) * prec_in[b * cN + nn] + wloc;
    }

    float ret[2] = {1.f, 1.f};
    for (int r = 0; r < cR; ++r) {
        const float mx  = blockReduceMax(fmaxf(sim[r][0], sim[r][1]), red);
        const float e0  = __expf(sim[r][0] - mx);
        const float e1  = __expf(sim[r][1] - mx);
        const float tot = blockReduceSum(e0 + e1, red);
#pragma unroll
        for (int k = 0; k < 2; ++k) {
            const int nn = nidx[k];
            const float content = ((k == 0) ? e0 : e1) / tot;
            float fs = 0.f, bs = 0.f;
#pragma unroll
            for (int p = 0; p < 16; ++p) {
                fs += fwd_part[((size_t)(b * 16 + p) * cR + r) * cN + nn];
                bs += bwd_part[((size_t)(b * 16 + p) * cR + r) * cN + nn];
            }
            const float rwval = m0[r] * fs + m1[r] * bs + m2[r] * content;
            o_rw[(size_t)(b * cR + r) * cN + nn] = rwval;
            ret[k] *= (1.f - fg[r] * rwval);
        }
    }
#pragma unroll
    for (int k = 0; k < 2; ++k) {
        const int nn = nidx[k];
        const float u = usage_in[b * cN + nn];
        const float w = wwv[b * cN + nn];
        o_usage[b * cN + nn] = (u + w - u * w) * ret[k];
    }
}

// ---------------------------------------------------------------------------
// K6: read_val[b,r,:] = sum_n rw[b,r,n] * mem[b,n,:].  grid = B*R, block=256
// ---------------------------------------------------------------------------
__global__ __launch_bounds__(256) void k_readval(const float* __restrict__ o_rw,
                                                 const float* __restrict__ mem,
                                                 float* __restrict__ o_rv) {
    __shared__ float s[cW];
    const int b = blockIdx.x >> 2, r = blockIdx.x & 3, tid = threadIdx.x;
    float acc[cW];
#pragma unroll
    for (int w = 0; w < cW; ++w) acc[w] = 0.f;
    for (int nn = tid; nn < cN; nn += 256) {
        const float wt = o_rw[(size_t)(b * cR + r) * cN + nn];
        const float4* mp = (const float4*)(mem + ((size_t)b * cN + nn) * cW);
#pragma unroll
        for (int q = 0; q < 16; ++q) {
            float4 m4 = mp[q];
            acc[4 * q + 0] += wt * m4.x;
            acc[4 * q + 1] += wt * m4.y;
            acc[4 * q + 2] += wt * m4.z;
            acc[4 * q + 3] += wt * m4.w;
        }
    }
    if (tid < cW) s[tid] = 0.f;
    __syncthreads();
#pragma unroll
    for (int w = 0; w < cW; ++w) atomicAdd(&s[w], acc[w]);
    __syncthreads();
    if (tid < cW) o_rv[(size_t)(b * cR + r) * cW + tid] = s[tid];
}

// ---------------------------------------------------------------------------
extern "C" void kernel_launch(void* const* d_in, const int* in_sizes, int n_in,
                              void* d_out, int out_size, void* d_ws, size_t ws_size,
                              hipStream_t stream) {
    (void)in_sizes; (void)n_in; (void)out_size; (void)ws_size;
    const float* wkey   = (const float*)d_in[0];
    const float* wstr   = (const float*)d_in[1];
    const float* agate  = (const float*)d_in[2];
    const float* wgate  = (const float*)d_in[3];
    const float* wvec   = (const float*)d_in[4];
    const float* erase  = (const float*)d_in[5];
    const float* rkeys  = (const float*)d_in[6];
    const float* rstr   = (const float*)d_in[7];
    const float* rmodes = (const float*)d_in[8];
    const float* fgates = (const float*)d_in[9];
    const float* memory = (const float*)d_in[10];
    const float* usage  = (const float*)d_in[11];
    const float* rw_in  = (const float*)d_in[13];   // read_weights (12 = write_weight, unused)
    const float* lnk_in = (const float*)d_in[14];
    const float* prec_i = (const float*)d_in[15];

    float* out = (float*)d_out;
    float* o_rv    = out;                                  // (B,R,W)   4096
    float* o_mem   = o_rv + (size_t)cB * cR * cW;          // (B,N,W)   2097152
    float* o_ww    = o_mem + (size_t)cB * cN * cW;         // (B,1,N)   32768
    float* o_rw    = o_ww + (size_t)cB * cN;               // (B,R,N)   131072
    float* o_link  = o_rw + (size_t)cB * cR * cN;          // (B,N,N)   67108864
    float* o_prec  = o_link + (size_t)cB * cN * cN;        // (B,1,N)   32768
    float* o_usage = o_prec + (size_t)cB * cN;             // (B,1,N)   32768

    float* ws       = (float*)d_ws;
    float* ws_alloc = ws;                                   // B*N
    float* ws_wwsum = ws_alloc + (size_t)cB * cN;           // B
    float* ws_fwd   = ws_wwsum + 32;                        // B*16*R*N
    float* ws_bwd   = ws_fwd + (size_t)cB * 16 * cR * cN;   // B*16*R*N

    k_alloc<<<cB, 1024, 0, stream>>>(usage, ws_alloc);
    k_wcontent<<<cB, 1024, 0, stream>>>(wkey, wstr, agate, wgate, memory,
                                        ws_alloc, o_ww, ws_wwsum);
    k_memupd<<<(cB * cN * 16) / 256, 256, 0, stream>>>(memory, erase, wvec, o_ww, o_mem);
    k_link<<<cB * 16 * 16, 256, 0, stream>>>(lnk_in, prec_i, rw_in, o_ww,
                                             o_link, ws_fwd, ws_bwd);
    k_read<<<cB, 1024, 0, stream>>>(rkeys, rstr, rmodes, fgates, usage, prec_i,
                                    o_mem, o_ww, ws_wwsum, ws_fwd, ws_bwd,
                                    o_rw, o_prec, o_usage);
    k_readval<<<cB * cR, 256, 0, stream>>>(o_rw, o_mem, o_rv);
}